// QOC_8220567405153
// MI455X (gfx1250) — compile-verified
//
#include <hip/hip_runtime.h>

// ---------------------------------------------------------------------------
// QOC propagator for MI455X (gfx1250):
//   U_t = expm(-i*(Hd + a0*Hc0 + a1*Hc1)*DT)  (Taylor-12 Horner, fp32 WMMA)
//   out = U_0 @ U_1 @ ... @ U_2047            (ordered chain reduction)
// All matmuls use V_WMMA_F32_16X16X4_F32 (fp32 in/accum) for precision over
// a 2048-term unitary product. Matrices live in LDS as re/im planes with a
// padded row stride (65) so column-wise A-fragment reads are conflict-free.
// ---------------------------------------------------------------------------

namespace {
constexpr int   kDim    = 64;
constexpr int   kLd     = 65;             // padded LDS row stride
constexpr int   kPlane  = kDim * kLd;     // floats per re/im plane
constexpr int   kTSteps = 2048;
constexpr float kDt     = 0.05f;
constexpr int   kTerms  = 12;             // Taylor order (||M|| <~ 0.3)
constexpr int   kMatF   = kDim * kDim * 2; // 8192 interleaved floats / matrix
}

typedef float v2f_t __attribute__((ext_vector_type(2)));
typedef float v8f_t __attribute__((ext_vector_type(8)));

#define WMMA_F32(A, B, C) \
  __builtin_amdgcn_wmma_f32_16x16x4_f32(false, (A), false, (B), (short)0, (C), false, false)

// One wave computes TWO 16x16 complex tiles of C = A @ B that share the same
// column block tn (tile rows tm0 and tm0+2), so B fragments (and the negated
// B_im) are loaded once per K-step and feed 8 WMMAs.
// Fragment layouts per CDNA5 ISA 7.12.2:
//   A (16x4, 2 VGPR):  lane L, vgpr j -> M = L%16, K = j + 2*(L/16)
//   B (4x16, 2 VGPR):  lane L, vgpr j -> K = j + 2*(L/16), N = L%16
//   C (16x16, 8 VGPR): lane L, vgpr i -> M = i + 8*(L/16), N = L%16
// Float WMMA only allows C-negate (NEG[1:0]==0), so B_im is negated in-reg.
__device__ __forceinline__ void wave_cmm_dual(const float* Are, const float* Aim,
                                              const float* Bre, const float* Bim,
                                              int tm0, int tn, int lane,
                                              v8f_t& cre0, v8f_t& cim0,
                                              v8f_t& cre1, v8f_t& cim1)
{
  const int h  = lane >> 4;        // 0 or 1
  const int ml = lane & 15;
  const int m0 = tm0 * 16 + ml;    // A row, tile 0
  const int m1 = m0 + 32;          // A row, tile 1 (tm0 + 2)
  const int nb = tn * 16 + ml;     // B/C column
  v8f_t r0 = {}, i0 = {}, r1 = {}, i1 = {};
#pragma unroll 4
  for (int k0 = 0; k0 < kDim; k0 += 4) {
    const int ka = k0 + 2 * h;
    v2f_t br  = { Bre[ka * kLd + nb], Bre[(ka + 1) * kLd + nb] };
    v2f_t bi  = { Bim[ka * kLd + nb], Bim[(ka + 1) * kLd + nb] };
    v2f_t nbi = { -bi.x, -bi.y };
    v2f_t ar0 = { Are[m0 * kLd + ka], Are[m0 * kLd + ka + 1] };
    v2f_t ai0 = { Aim[m0 * kLd + ka], Aim[m0 * kLd + ka + 1] };
    v2f_t ar1 = { Are[m1 * kLd + ka], Are[m1 * kLd + ka + 1] };
    v2f_t ai1 = { Aim[m1 * kLd + ka], Aim[m1 * kLd + ka + 1] };
    // group same-B WMMAs adjacently (operand reuse in the matrix pipe)
    r0 = WMMA_F32(ar0, br,  r0);       // + Ar*Br
    r1 = WMMA_F32(ar1, br,  r1);
    i0 = WMMA_F32(ai0, br,  i0);       // + Ai*Br
    i1 = WMMA_F32(ai1, br,  i1);
    i0 = WMMA_F32(ar0, bi,  i0);       // + Ar*Bi
    i1 = WMMA_F32(ar1, bi,  i1);
    r0 = WMMA_F32(ai0, nbi, r0);       // - Ai*Bi
    r1 = WMMA_F32(ai1, nbi, r1);
  }
  cre0 = r0; cim0 = i0; cre1 = r1; cim1 = i1;
}

// Workgroup-level complex matmul: D = scale*(A @ B) (+ I if addI).
// 8 waves x 2 tiles = 16 tiles. Accumulate in registers, barrier, write back:
// D may alias A or B.
__device__ __forceinline__ void wg_cmm(const float* Are, const float* Aim,
                                       const float* Bre, const float* Bim,
                                       float* Dre, float* Dim,
                                       float scale, bool addI,
                                       int wave, int lane)
{
  const int tm0 = wave >> 2;       // 0..1 ; second tile at tm0+2
  const int tn  = wave & 3;
  v8f_t cre0, cim0, cre1, cim1;
  wave_cmm_dual(Are, Aim, Bre, Bim, tm0, tn, lane, cre0, cim0, cre1, cim1);
  __syncthreads();   // all reads of A/B done before overwriting D (may alias)
  const int h  = lane >> 4;
  const int ml = lane & 15;
  const int c  = tn * 16 + ml;
#pragma unroll
  for (int i = 0; i < 8; ++i) {
    const int r0 = tm0 * 16 + i + 8 * h;
    const int r1 = r0 + 32;
    Dre[r0 * kLd + c] = cre0[i] * scale + ((addI && r0 == c) ? 1.0f : 0.0f);
    Dim[r0 * kLd + c] = cim0[i] * scale;
    Dre[r1 * kLd + c] = cre1[i] * scale + ((addI && r1 == c) ? 1.0f : 0.0f);
    Dim[r1 * kLd + c] = cim1[i] * scale;
  }
  __syncthreads();   // D visible to all waves
}

// Stage 1: workgroup b computes P_b = U_{b*chunk} @ ... @ U_{b*chunk+chunk-1}.
__global__ void __launch_bounds__(256)
qoc_expm_chunk_kernel(const float* __restrict__ a,   // (2, 2048) f32
                      const float* __restrict__ Hd,  // (64,64) c64 interleaved
                      const float* __restrict__ Hc,  // (2,64,64) c64 interleaved
                      float* __restrict__ out,       // gridDim.x matrices, c64
                      int chunk)
{
  __shared__ float lds[6 * kPlane];   // ~97.5 KB (<< 320 KB WGP LDS)
  float* Mre = lds + 0 * kPlane;
  float* Mim = lds + 1 * kPlane;
  float* Sre = lds + 2 * kPlane;
  float* Sim = lds + 3 * kPlane;
  float* Pre = lds + 4 * kPlane;
  float* Pim = lds + 5 * kPlane;

  const int tid  = threadIdx.x;
  const int wave = tid >> 5;
  const int lane = tid & 31;
  const int t0   = blockIdx.x * chunk;

  for (int s = 0; s < chunk; ++s) {
    const int t  = t0 + s;
    const float a0 = a[t];
    const float a1 = a[kTSteps + t];

    // Build M = -i*H*DT and S = I.  H = Hd + a0*Hc0 + a1*Hc1 (complex).
    // -i*(x+iy)*DT = (y - i*x)*DT  ->  Mre = DT*him, Mim = -DT*hre.
    for (int e = tid; e < kDim * kDim; e += 256) {
      const int r = e >> 6, c = e & 63;
      const float hre = Hd[2 * e]     + a0 * Hc[2 * e]     + a1 * Hc[2 * (4096 + e)];
      const float him = Hd[2 * e + 1] + a0 * Hc[2 * e + 1] + a1 * Hc[2 * (4096 + e) + 1];
      Mre[r * kLd + c] = kDt * him;
      Mim[r * kLd + c] = -kDt * hre;
      Sre[r * kLd + c] = (r == c) ? 1.0f : 0.0f;
      Sim[r * kLd + c] = 0.0f;
    }
    __syncthreads();

    // Horner: S <- I + (M @ S)/j, j = kTerms..1  =>  S = expm(M)
    for (int j = kTerms; j >= 1; --j) {
      wg_cmm(Mre, Mim, Sre, Sim, Sre, Sim, 1.0f / (float)j, true, wave, lane);
    }

    // Fold into running product: P <- P @ U_t (order matters).
    if (s == 0) {
      for (int e = tid; e < kDim * kDim; e += 256) {
        const int r = e >> 6, c = e & 63;
        Pre[r * kLd + c] = Sre[r * kLd + c];
        Pim[r * kLd + c] = Sim[r * kLd + c];
      }
      __syncthreads();
    } else {
      wg_cmm(Pre, Pim, Sre, Sim, Pre, Pim, 1.0f, false, wave, lane);
    }
  }

  // Store partial product (interleaved complex64).
  float* dst = out + (size_t)blockIdx.x * kMatF;
  for (int e = tid; e < kDim * kDim; e += 256) {
    const int r = e >> 6, c = e & 63;
    dst[2 * e]     = Pre[r * kLd + c];
    dst[2 * e + 1] = Pim[r * kLd + c];
  }
}

// Reduction: workgroup g computes dst[g] = src[g*chunk] @ ... @ src[g*chunk+chunk-1].
__global__ void __launch_bounds__(256)
qoc_chain_kernel(const float* __restrict__ src, float* __restrict__ dst, int chunk)
{
  __shared__ float lds[4 * kPlane];   // ~65 KB
  float* Pre = lds + 0 * kPlane;
  float* Pim = lds + 1 * kPlane;
  float* Sre = lds + 2 * kPlane;
  float* Sim = lds + 3 * kPlane;

  const int tid  = threadIdx.x;
  const int wave = tid >> 5;
  const int lane = tid & 31;
  const size_t base = (size_t)blockIdx.x * chunk;

  // P = src[base]
  {
    const float* m0 = src + base * kMatF;
    for (int e = tid; e < kDim * kDim; e += 256) {
      const int r = e >> 6, c = e & 63;
      Pre[r * kLd + c] = m0[2 * e];
      Pim[r * kLd + c] = m0[2 * e + 1];
    }
  }
  __syncthreads();

  for (int i = 1; i < chunk; ++i) {
    const float* mi = src + (base + i) * kMatF;
    for (int e = tid; e < kDim * kDim; e += 256) {
      const int r = e >> 6, c = e & 63;
      Sre[r * kLd + c] = mi[2 * e];
      Sim[r * kLd + c] = mi[2 * e + 1];
    }
    __syncthreads();
    wg_cmm(Pre, Pim, Sre, Sim, Pre, Pim, 1.0f, false, wave, lane);
  }

  float* d = dst + (size_t)blockIdx.x * kMatF;
  for (int e = tid; e < kDim * kDim; e += 256) {
    const int r = e >> 6, c = e & 63;
    d[2 * e]     = Pre[r * kLd + c];
    d[2 * e + 1] = Pim[r * kLd + c];
  }
}

extern "C" void kernel_launch(void* const* d_in, const int* in_sizes, int n_in,
                              void* d_out, int out_size, void* d_ws, size_t ws_size,
                              hipStream_t stream) {
  (void)in_sizes; (void)n_in; (void)out_size;
  const float* a  = (const float*)d_in[0];   // (2,2048) f32
  const float* Hd = (const float*)d_in[1];   // (64,64) c64 -> interleaved f32
  const float* Hc = (const float*)d_in[2];   // (2,64,64) c64 -> interleaved f32

  // Choose stage-1 chunk from available scratch (deterministic: ws_size is
  // fixed). Prefer small chunks (short serial chains, more workgroups).
  int chunk1 = 4;                            // 512 workgroups, 17.1 MB scratch
  size_t need;
  for (;;) {
    const size_t n1 = (size_t)(kTSteps / chunk1);
    need = (n1 + n1 / 16 + 1) * (size_t)kMatF * sizeof(float);
    if (need <= ws_size || chunk1 >= kTSteps) break;
    chunk1 <<= 1;
  }
  const int n1 = kTSteps / chunk1;

  float* buf0 = (float*)d_ws;                          // n1 matrices
  float* buf1 = buf0 + (size_t)n1 * kMatF;             // <= n1/16 matrices

  qoc_expm_chunk_kernel<<<n1, 256, 0, stream>>>(
      a, Hd, Hc, (n1 == 1) ? (float*)d_out : buf0, chunk1);

  // Factor-16 ordered chain reduction, ping-ponging buffers; final level
  // writes straight to d_out.
  float* src = buf0;
  float* dst = buf1;
  int n = n1;
  while (n > 1) {
    const int chunk = (n >= 16) ? 16 : n;
    const int ng = n / chunk;                // n is a power of two
    if (ng == 1) {
      qoc_chain_kernel<<<1, 256, 0, stream>>>(src, (float*)d_out, chunk);
    } else {
      qoc_chain_kernel<<<ng, 256, 0, stream>>>(src, dst, chunk);
      float* t = src; src = dst; dst = t;
    }
    n = ng;
  }
}